// DipoleLayer_44839458570528
// MI455X (gfx1250) — compile-verified
//
#include <hip/hip_runtime.h>

// ---------------------------------------------------------------------------
// DipoleLayer for MI455X (gfx1250, wave32).
// Kernel 1: fused MLP — two 128x128 GEMMs (V_WMMA_F32_16X16X4_F32) with the
//           shifted-softplus intermediate held in LDS (bank-conflict padded).
// Kernel 2: cosine-cutoff-weighted gather reduction; neighbor data packed as
//           float4 in LDS (single ds_load_b128/j), scalar (saddr) q gather.
// ---------------------------------------------------------------------------

typedef float v2f __attribute__((ext_vector_type(2)));
typedef float v8f __attribute__((ext_vector_type(8)));

#define FDIM   128
#define NNB    255
#define AATOMS 256
#define BBATCH 4
#define LDSPAD 132   // 128 + 4 dwords: lane i hits bank (4i+k)%64 -> conflict-free

__device__ __forceinline__ float shifted_softplus(float x) {
    // softplus(x) - ln(2), numerically stable
    return fmaxf(x, 0.0f) + log1pf(__expf(-fabsf(x))) - 0.69314718055994530942f;
}

// Y[1024 x 128] = ssp( ssp(X @ W1 + b1) @ W2 + b2 )
// One block (8 waves) per 16-row strip; intermediate strip lives in LDS.
__global__ void __launch_bounds__(256)
mlp2_ssp_wmma(const float* __restrict__ X,
              const float* __restrict__ W1,
              const float* __restrict__ b1,
              const float* __restrict__ W2,
              const float* __restrict__ b2,
              float* __restrict__ Y)
{
    __shared__ float q1s[16][LDSPAD];

    const int wv   = threadIdx.x >> 5;   // wave id in block: 0..7 -> N tile
    const int lane = threadIdx.x & 31;
    const int half = lane >> 4;          // 0: K pair (0,1) ; 1: K pair (2,3)
    const int lr   = lane & 15;          // row (A) / col (B) within tile

    const int tm = blockIdx.x * 16;      // row strip base
    const int tn = wv * 16;              // col tile base
    const int n  = tn + lr;

    // ---- stage 1: q1 = ssp(X @ W1 + b1), result tile -> LDS ----
    {
        const float* xrow = X  + (size_t)(tm + lr) * FDIM;
        const float* wcol = W1 + n;
        v8f c = {};
#pragma unroll
        for (int k = 0; k < FDIM; k += 4) {
            v2f a, b;
            a.x = xrow[k + 2 * half];
            a.y = xrow[k + 2 * half + 1];
            b.x = wcol[(size_t)(k + 2 * half) * FDIM];
            b.y = wcol[(size_t)(k + 2 * half + 1) * FDIM];
            c = __builtin_amdgcn_wmma_f32_16x16x4_f32(false, a, false, b,
                                                      (short)0, c, false, false);
        }
        const float bn = b1[n];
#pragma unroll
        for (int r = 0; r < 8; ++r)
            q1s[r + 8 * half][n] = shifted_softplus(c[r] + bn);
    }
    __syncthreads();

    // ---- stage 2: q2 = ssp(q1 @ W2 + b2), A from LDS ----
    {
        const float* q1r  = &q1s[lr][0];
        const float* wcol = W2 + n;
        v8f c = {};
#pragma unroll
        for (int k = 0; k < FDIM; k += 4) {
            v2f a, b;
            a.x = q1r[k + 2 * half];
            a.y = q1r[k + 2 * half + 1];
            b.x = wcol[(size_t)(k + 2 * half) * FDIM];
            b.y = wcol[(size_t)(k + 2 * half + 1) * FDIM];
            c = __builtin_amdgcn_wmma_f32_16x16x4_f32(false, a, false, b,
                                                      (short)0, c, false, false);
        }
        const float bn = b2[n];
#pragma unroll
        for (int r = 0; r < 8; ++r)
            Y[(size_t)(tm + r + 8 * half) * FDIM + n] = shifted_softplus(c[r] + bn);
    }
}

// mu[b,i,f,0..2] = sum_j q[b, nbr[b,i,j], f] * w[b,i,j] * v[b,i,j,0..2]
// w = mask * 0.5*(cos(pi*r/5)+1) * (r < 5)
__global__ void __launch_bounds__(FDIM)
dipole_kernel(const float*     __restrict__ q,      // [B*A, 128]
              const float*     __restrict__ r_ij,   // [B, A, N]
              const float*     __restrict__ v_ij,   // [B, A, N, 3]
              const long long* __restrict__ nbr,    // [B, A, N] (int64)
              const float*     __restrict__ mask,   // [B, A, N]
              float*           __restrict__ mu)     // [B, A, 128, 3]
{
    const int i = blockIdx.x;          // atom
    const int b = blockIdx.y;          // batch
    const int f = threadIdx.x;         // feature 0..127

    const size_t pair = (size_t)(b * AATOMS + i) * NNB;
    const char*  qb   = (const char*)(q + (size_t)b * AATOMS * FDIM);

    // {w*v0, w*v1, w*v2, byte offset of neighbor row} -> one ds_load_b128 per j
    __shared__ float4 spack[FDIM];

    float acc0 = 0.0f, acc1 = 0.0f, acc2 = 0.0f;

    for (int base = 0; base < NNB; base += FDIM) {
        const int j = base + f;
        if (j < NNB) {
            const float r  = r_ij[pair + j];
            const float cc = 0.5f * (__cosf(0.62831853071795864769f * r) + 1.0f); // pi/5
            const float w  = (r < 5.0f ? cc : 0.0f) * mask[pair + j];
            const int  off = ((int)nbr[pair + j]) << 9;   // nbr * 128 * 4 bytes
            float4 t;
            t.x = w * v_ij[(pair + j) * 3 + 0];
            t.y = w * v_ij[(pair + j) * 3 + 1];
            t.z = w * v_ij[(pair + j) * 3 + 2];
            t.w = __int_as_float(off);
            spack[f] = t;
        }
        __syncthreads();

        const int cnt = min(FDIM, NNB - base);
#pragma unroll 4
        for (int jj = 0; jj < cnt; ++jj) {
            const float4 t = spack[jj];
            // neighbor index is uniform across the block: force a scalar base
            // so the gather lowers to global_load_b32 v, voff, s[base]
            const int off = __builtin_amdgcn_readfirstlane(__float_as_int(t.w));
            const float qv = *(const float*)(qb + off + (size_t)f * 4);
            acc0 = fmaf(qv, t.x, acc0);
            acc1 = fmaf(qv, t.y, acc1);
            acc2 = fmaf(qv, t.z, acc2);
        }
        __syncthreads();
    }

    float* out = mu + ((size_t)(b * AATOMS + i) * FDIM + f) * 3;
    out[0] = acc0;
    out[1] = acc1;
    out[2] = acc2;
}

extern "C" void kernel_launch(void* const* d_in, const int* in_sizes, int n_in,
                              void* d_out, int out_size, void* d_ws, size_t ws_size,
                              hipStream_t stream)
{
    const float*     x     = (const float*)d_in[0];      // [4,256,128]
    const float*     r_ij  = (const float*)d_in[1];      // [4,256,255]
    const float*     v_ij  = (const float*)d_in[2];      // [4,256,255,3]
    const long long* nb    = (const long long*)d_in[3];  // [4,256,255] int64
    const float*     mask  = (const float*)d_in[4];      // [4,256,255]
    const float*     W1    = (const float*)d_in[5];      // [128,128]
    const float*     b1    = (const float*)d_in[6];      // [128]
    const float*     W2    = (const float*)d_in[7];      // [128,128]
    const float*     b2    = (const float*)d_in[8];      // [128]

    float* q2 = (float*)d_ws;                            // [1024,128] = 512 KB

    // 1024 rows / 16-row strips -> 64 blocks x 8 waves (both GEMMs fused)
    mlp2_ssp_wmma<<<64, 256, 0, stream>>>(x, W1, b1, W2, b2, q2);

    dipole_kernel<<<dim3(AATOMS, BBATCH), FDIM, 0, stream>>>(
        q2, r_ij, v_ij, nb, mask, (float*)d_out);
}